// MatchLayer_31121333027528
// MI455X (gfx1250) — compile-verified
//
#include <hip/hip_runtime.h>
#include <stdint.h>

// ---- problem constants (match reference) ----
#define N_ROWS    2000000
#define N_FEAT    64
#define PAT       16

// ---- tiling ----
#define TILE_ROWS 128
#define TILE_FLTS (TILE_ROWS * N_FEAT)          // 8192 floats
#define TILE_BYTES (TILE_FLTS * 4)              // 32768 B
#define N_TILES   (N_ROWS / TILE_ROWS)          // 15625 (exact)
#define THREADS   256
#define BLOCKS    3125                          // 15625 / 3125 = 5 tiles per block

// Issue 8 async 16B/lane copies: one tile (32KB) per 256-thread block.
// CDNA5 ISA: both addresses take the instruction offset
// -> the offset: immediate advances BOTH the global and LDS addresses.
// th:TH_LOAD_NT: x is streamed exactly once (512MB > 192MB L2) -> don't pollute L2.
#define ASYNC_COPY_TILE(lOff, gOff, sBase)                                          \
    asm volatile(                                                                   \
        "global_load_async_to_lds_b128 %0, %1, %2 offset:0 th:TH_LOAD_NT\n\t"       \
        "global_load_async_to_lds_b128 %0, %1, %2 offset:4096 th:TH_LOAD_NT\n\t"    \
        "global_load_async_to_lds_b128 %0, %1, %2 offset:8192 th:TH_LOAD_NT\n\t"    \
        "global_load_async_to_lds_b128 %0, %1, %2 offset:12288 th:TH_LOAD_NT\n\t"   \
        "global_load_async_to_lds_b128 %0, %1, %2 offset:16384 th:TH_LOAD_NT\n\t"   \
        "global_load_async_to_lds_b128 %0, %1, %2 offset:20480 th:TH_LOAD_NT\n\t"   \
        "global_load_async_to_lds_b128 %0, %1, %2 offset:24576 th:TH_LOAD_NT\n\t"   \
        "global_load_async_to_lds_b128 %0, %1, %2 offset:28672 th:TH_LOAD_NT\n\t"   \
        :: "v"(lOff), "v"(gOff), "s"(sBase) : "memory")

__global__ __launch_bounds__(THREADS)
void match_stream_kernel(const float* __restrict__ x,
                         const float* __restrict__ thr,
                         const int*   __restrict__ pat,
                         float*       __restrict__ out) {
    __shared__ __align__(16) float smem[2][TILE_FLTS];   // 64 KB double buffer

    const int tid  = threadIdx.x;
    const int lane = tid & 31;
    const int wave = tid >> 5;

    // Each half-wave lane owns 4 consecutive features: f0..f0+3.
    const int f0 = (lane & 15) * 4;
    const float4 t4 = *(const float4*)(thr + f0);

    // 4-bit mask of owned features that appear in pat_index.
    unsigned need = 0u;
#pragma unroll
    for (int i = 0; i < PAT; ++i) {
        unsigned d = (unsigned)(pat[i] - f0);
        if (d < 4u) need |= 1u << d;
    }

    const uint64_t xbase = (uint64_t)(uintptr_t)x;
    const uint32_t ldsA  = (uint32_t)(uintptr_t)&smem[0][0];  // low 32 bits = LDS offset
    const uint32_t ldsB  = (uint32_t)(uintptr_t)&smem[1][0];
    const uint32_t lane16 = (uint32_t)tid * 16u;

    const int stride = (int)gridDim.x;
    int t   = (int)blockIdx.x;
    int cur = 0;

    // Prologue: prefetch first tile into buffer 0.
    if (t < N_TILES) {
        ASYNC_COPY_TILE(ldsA + lane16, (uint32_t)t * (uint32_t)TILE_BYTES + lane16, xbase);
    }

    for (; t < N_TILES; t += stride) {
        const int tn = t + stride;
        if (tn < N_TILES) {
            // Prefetch next tile into the other buffer, then wait for current
            // tile's 8 in-order async loads (<= 8 outstanding remain).
            const uint32_t lNext = (cur ? ldsA : ldsB) + lane16;
            ASYNC_COPY_TILE(lNext, (uint32_t)tn * (uint32_t)TILE_BYTES + lane16, xbase);
            asm volatile("s_wait_asynccnt 0x8" ::: "memory");
        } else {
            asm volatile("s_wait_asynccnt 0x0" ::: "memory");
        }
        __syncthreads();   // whole tile t visible in LDS to all waves

        // ---- consume tile t: wave handles 16 rows, 2 rows per ds_load_b128 ----
        const float* buf = &smem[cur][0];
        const int rowBase = wave * 16;           // within tile
        float res = 0.0f;                        // result for row rowBase+lane (lane<16)
#pragma unroll
        for (int i = 0; i < 8; ++i) {
            const int row = rowBase + 2 * i + (lane >> 4);   // lanes 0-15: even row, 16-31: odd row
            const float4 v = *(const float4*)(buf + row * N_FEAT + f0);
            unsigned pass = (v.x > t4.x ? 1u : 0u) |
                            (v.y > t4.y ? 2u : 0u) |
                            (v.z > t4.z ? 4u : 0u) |
                            (v.w > t4.w ? 8u : 0u);
            const bool ok = (pass & need) == need;
            const unsigned b = (unsigned)__ballot(ok);       // wave32: low 32 bits
            if ((lane >> 1) == i) {
                const unsigned half = (lane & 1) ? (b >> 16) : (b & 0xFFFFu);
                res = (half == 0xFFFFu) ? 1.0f : 0.0f;
            }
        }
        if (lane < 16) {
            // 8MB of write-once results: non-temporal, coalesced 64B per wave.
            __builtin_nontemporal_store(res, &out[t * TILE_ROWS + rowBase + lane]);
        }

        __syncthreads();   // all reads of smem[cur] done before it is refilled next iter
        cur ^= 1;
    }
}

extern "C" void kernel_launch(void* const* d_in, const int* in_sizes, int n_in,
                              void* d_out, int out_size, void* d_ws, size_t ws_size,
                              hipStream_t stream) {
    const float* x   = (const float*)d_in[0];
    const float* thr = (const float*)d_in[1];
    const int*   pat = (const int*)d_in[2];
    float*       out = (float*)d_out;
    (void)in_sizes; (void)n_in; (void)out_size; (void)d_ws; (void)ws_size;
    match_stream_kernel<<<BLOCKS, THREADS, 0, stream>>>(x, thr, pat, out);
}